// SelfGeometricStructureEmbedding_67748814127203
// MI455X (gfx1250) — compile-verified
//
#include <hip/hip_runtime.h>

typedef __attribute__((ext_vector_type(2))) float        v2f;
typedef __attribute__((ext_vector_type(8))) float        v8f;
typedef __attribute__((ext_vector_type(4))) unsigned int u32x4;
typedef __attribute__((ext_vector_type(8))) unsigned int u32x8;

#define NPTS        1024
#define HID         128
#define LDS_STRIDE  132                 // 128 dwords + 4 pad dwords: 16B-aligned rows, conflict-free b64 reads
#define INV_SIGMA   5.0f                // 1 / 0.2
#define FREQ_STEP   0.7498942093324559f // 10000^(-1/32)
#define FREQ_HALF   0.8659643233600653f // 10000^(-1/64)

// out[(n*1024+m)*128 + o] = sum_h emb[h] * W[o][h] + b[o]
// emb[2i] = sin(d * 10000^(-i/64)), emb[2i+1] = cos(...), d = |p_n - p_m| / 0.2
__global__ __launch_bounds__(256) void geo_embed_wmma(
    const float* __restrict__ points,   // (1024, 3)
    const float* __restrict__ W,        // (128, 128) row-major (out, in)
    const float* __restrict__ bias,     // (128,)
    float* __restrict__ out,            // (1024*1024, 128)
    int num_tiles)                      // 65536 tiles of 16 pairs
{
  __shared__ float ldsW[HID * LDS_STRIDE];

  const int tid  = threadIdx.x;
  const int lane = tid & 31;
  const int wave = tid >> 5;     // 8 waves per block (wave32)
  const int hi   = lane >> 4;    // 0: lanes 0-15, 1: lanes 16-31
  const int l16  = lane & 15;

  // ---- Stage W (128x128 f32, 64KB) into LDS via TDM, row-padded to LDS_STRIDE dwords ----
#if defined(__gfx1250__)
  if (wave == 0) {
    unsigned long long ga = (unsigned long long)(uintptr_t)W;
    unsigned int lb = (unsigned int)(uintptr_t)ldsW;     // low 32 bits of generic addr = LDS offset
    u32x4 g0; u32x8 g1; u32x4 g2; u32x4 g3;
    // D# group0: count=1 (valid user descriptor), lds_addr, 57-bit global addr, type=2 ("image")
    g0[0] = 1u;
    g0[1] = lb;
    g0[2] = (unsigned int)ga;
    g0[3] = (unsigned int)((ga >> 32) & 0x01FFFFFFu) | (2u << 30);
    // D# group1: data_size=4B (code 2), pad_enable=1, pad_interval=6 (128 dwords),
    //            pad_amount=3 (4 dwords) -> effective LDS row stride = 132 dwords
    g1[0] = (2u << 16) | (1u << 20) | (6u << 22) | (3u << 25);
    g1[1] = (128u << 16);   // tensor_dim0 = 128  (bits 79:48)
    g1[2] = (128u << 16);   // tensor_dim1 = 128  (bits 111:80)
    g1[3] = (128u << 16);   // tile_dim0   = 128  (bits 127:112)
    g1[4] = 128u;           // tile_dim1 = 128, tile_dim2 = 0
    g1[5] = 128u;           // tensor_dim0_stride = 128
    g1[6] = 0u;
    g1[7] = 0u;
    g2[0] = g2[1] = g2[2] = g2[3] = 0u;   // dims 2..4 unused (2D tensor)
    g3[0] = g3[1] = g3[2] = g3[3] = 0u;
    asm volatile("tensor_load_to_lds %0, %1, %2, %3"
                 :: "s"(g0), "s"(g1), "s"(g2), "s"(g3) : "memory");
    __builtin_amdgcn_s_wait_tensorcnt(0);
  }
#else
  // non-gfx1250 / host-pass fallback: cooperative staged copy
  for (int idx = tid; idx < HID * HID; idx += 256)
    ldsW[(idx >> 7) * LDS_STRIDE + (idx & 127)] = W[idx];
#endif
  __syncthreads();

  // Bias per lane for all 8 output tiles (C/D layout: col N = l16 in every acc VGPR)
  float bv[8];
  #pragma unroll
  for (int o = 0; o < 8; ++o) bv[o] = bias[o * 16 + l16];

  const int waves_total = gridDim.x * 8;
  for (int tile = blockIdx.x * 8 + wave; tile < num_tiles; tile += waves_total) {
    // This lane owns pair p (both lane halves cover the same 16 pairs; halves differ in K).
    const int p = tile * 16 + l16;
    const int n = p >> 10;
    const int m = p & (NPTS - 1);
    const float dx = points[3 * n + 0] - points[3 * m + 0];
    const float dy = points[3 * n + 1] - points[3 * m + 1];
    const float dz = points[3 * n + 2] - points[3 * m + 2];
    const float d  = sqrtf(dx * dx + dy * dy + dz * dz) * INV_SIGMA;

    // 8 accumulators (one per 16-wide output tile) stay live; A fragments are transient.
    v8f acc[8];
    #pragma unroll
    for (int o = 0; o < 8; ++o) {
      const float b0 = bv[o];
      acc[o] = (v8f){ b0, b0, b0, b0, b0, b0, b0, b0 };
    }

    // A fragment in WMMA 16x4 f32 layout: lanes 0-15 VGPR{0,1} = K{0,1},
    // lanes 16-31 = K{2,3}. K = 4*ks + 2*hi + {0,1} -> (sin,cos) of omega_i, i = 2*ks + hi.
    float omega = hi ? d * FREQ_HALF : d;
    const float* wbase = &ldsW[l16 * LDS_STRIDE + 2 * hi];
    #pragma unroll 4
    for (int ks = 0; ks < 32; ++ks) {
      v2f a;
      a[0] = __sinf(omega);
      a[1] = __cosf(omega);
      omega *= FREQ_STEP;
      const float* wp = wbase + 4 * ks;
      #pragma unroll
      for (int o = 0; o < 8; ++o) {
        // B fragment (4x16, K x N): lane n=l16 reads W[o*16+l16][4*ks + 2*hi + {0,1}]
        v2f bf = *(const v2f*)(wp + o * (16 * LDS_STRIDE));  // ds_load_b64, conflict-free
        acc[o] = __builtin_amdgcn_wmma_f32_16x16x4_f32(
                     false, a, false, bf, (short)0, acc[o], false, false);
      }
    }

    // D 16x16: VGPR v -> pair row M = v + 8*hi, col = o*16 + l16. NT stores (512MB >> L2).
    const size_t rbase = (size_t)(tile * 16 + 8 * hi) * HID + (size_t)l16;
    #pragma unroll
    for (int o = 0; o < 8; ++o) {
      const size_t base = rbase + (size_t)(o * 16);
      #pragma unroll
      for (int v = 0; v < 8; ++v)
        __builtin_nontemporal_store(acc[o][v], &out[base + (size_t)v * HID]);
    }
  }
}

extern "C" void kernel_launch(void* const* d_in, const int* in_sizes, int n_in,
                              void* d_out, int out_size, void* d_ws, size_t ws_size,
                              hipStream_t stream) {
  const float* points = (const float*)d_in[0];
  const float* W      = (const float*)d_in[1];
  const float* b      = (const float*)d_in[2];
  float* out          = (float*)d_out;

  const int num_tiles = (NPTS * NPTS) / 16;  // 65536
  dim3 grid(2048), block(256);               // grid-stride: 4 tiles per wave
  hipLaunchKernelGGL(geo_embed_wmma, grid, block, 0, stream,
                     points, W, b, out, num_tiles);
}